// GCN_30571577213137
// MI455X (gfx1250) — compile-verified
//
#include <hip/hip_runtime.h>

#define N_NODES   100000
#define D_FEAT    128
#define HIDDEN    128
#define N_CLASSES 10
#define N_GRAPHS  64

typedef __attribute__((ext_vector_type(2))) float v2f;
typedef __attribute__((ext_vector_type(8))) float v8f;

// ---------------------------------------------------------------------------
// Zero the accumulation workspace (agg + pooled sums + counts) every call.
// ---------------------------------------------------------------------------
__global__ void zero_ws_kernel(float* __restrict__ p, long n) {
    long i = (long)blockIdx.x * blockDim.x + threadIdx.x;
    long stride = (long)gridDim.x * blockDim.x;
    for (; i < n; i += stride) p[i] = 0.0f;
}

// ---------------------------------------------------------------------------
// Edge scatter-add: one wave per edge. Each lane loads 16B (float4) of the
// source row (512B coalesced per wave) and atomically adds into agg[dst].
// x and agg are L2-resident (51.2MB each << 192MB L2).
// ---------------------------------------------------------------------------
__global__ void scatter_add_kernel(const float* __restrict__ x,
                                   const int* __restrict__ ei,
                                   float* __restrict__ agg, int n_edges) {
    int wid  = (int)(((long)blockIdx.x * blockDim.x + threadIdx.x) >> 5);
    int lane = threadIdx.x & 31;
    if (wid >= n_edges) return;
    int src = ei[wid];
    int dst = ei[n_edges + wid];
    const float4 v = *(const float4*)(x + (size_t)src * D_FEAT + lane * 4);
    float* a = agg + (size_t)dst * D_FEAT + lane * 4;
    atomicAdd(a + 0, v.x);
    atomicAdd(a + 1, v.y);
    atomicAdd(a + 2, v.z);
    atomicAdd(a + 3, v.w);
}

// ---------------------------------------------------------------------------
// Per-graph node counts (for mean pooling).
// ---------------------------------------------------------------------------
__global__ void count_kernel(const int* __restrict__ batch,
                             float* __restrict__ counts, int n) {
    int i = blockIdx.x * blockDim.x + threadIdx.x;
    if (i < n) atomicAdd(&counts[batch[i]], 1.0f);
}

// ---------------------------------------------------------------------------
// Fused GEMM + pooling.
//   h = agg @ w_rel^T + b_rel + x @ w_root^T        (never materialized)
//   summed[batch[m]] += h[m]                        (fused epilogue)
// Block = 256 threads = 8 waves; block owns a 16-node strip (LDS-staged),
// wave w owns output columns [16w, 16w+16). Each wave runs a K-loop of
// 32 steps with two V_WMMA_F32_16X16X4_F32 per step (full fp32 precision).
// ---------------------------------------------------------------------------
__global__ __launch_bounds__(256) void gemm_pool_kernel(
    const float* __restrict__ x, const float* __restrict__ agg,
    const int* __restrict__ batch,
    const float* __restrict__ w_rel, const float* __restrict__ b_rel,
    const float* __restrict__ w_root,
    float* __restrict__ summed) {

    __shared__ float sAgg[16 * D_FEAT];
    __shared__ float sX[16 * D_FEAT];

    const int tid   = threadIdx.x;
    const int mBase = blockIdx.x * 16;

    // Cooperative stage of the 16x128 A-strips (8KB each as float4s).
    const float4* gAgg = (const float4*)(agg + (size_t)mBase * D_FEAT);
    const float4* gX   = (const float4*)(x   + (size_t)mBase * D_FEAT);
    for (int i = tid; i < (16 * D_FEAT) / 4; i += 256) {
        ((float4*)sAgg)[i] = gAgg[i];
        ((float4*)sX)[i]   = gX[i];
    }
    __syncthreads();

    const int lane  = tid & 31;
    const int wave  = tid >> 5;
    const int nBase = wave * 16;        // this wave's output column base
    const int r     = lane & 15;        // A row / B column within tile
    const int kk    = (lane >> 4) * 2;  // upper half-wave holds K+2,K+3

    v8f c = {};

#if __has_builtin(__builtin_amdgcn_wmma_f32_16x16x4_f32)
    for (int k0 = 0; k0 < D_FEAT; k0 += 4) {
        v2f aA, aX, bR, bO;
        // A fragment (16x4 f32): lane holds row r, K = k0+kk, k0+kk+1
        aA.x = sAgg[r * D_FEAT + k0 + kk];
        aA.y = sAgg[r * D_FEAT + k0 + kk + 1];
        aX.x = sX[r * D_FEAT + k0 + kk];
        aX.y = sX[r * D_FEAT + k0 + kk + 1];
        // B fragment (4x16 f32): B[k][n] = w[n][k]  (w is row-major [H][F])
        bR.x = w_rel[(size_t)(nBase + r) * D_FEAT + k0 + kk];
        bR.y = w_rel[(size_t)(nBase + r) * D_FEAT + k0 + kk + 1];
        bO.x = w_root[(size_t)(nBase + r) * D_FEAT + k0 + kk];
        bO.y = w_root[(size_t)(nBase + r) * D_FEAT + k0 + kk + 1];
        c = __builtin_amdgcn_wmma_f32_16x16x4_f32(false, aA, false, bR,
                                                  (short)0, c, false, false);
        c = __builtin_amdgcn_wmma_f32_16x16x4_f32(false, aX, false, bO,
                                                  (short)0, c, false, false);
    }
#else
    // Scalar fallback producing the identical C/D fragment layout.
    {
        const int mOff = (lane >> 4) * 8;
        for (int v = 0; v < 8; ++v) {
            const int m = v + mOff;
            float acc = 0.0f;
            for (int k = 0; k < D_FEAT; ++k) {
                acc += sAgg[m * D_FEAT + k] * w_rel[(size_t)(nBase + r) * D_FEAT + k];
                acc += sX[m * D_FEAT + k]   * w_root[(size_t)(nBase + r) * D_FEAT + k];
            }
            c[v] = acc;
        }
    }
#endif

    // Bias (C/D layout: column n = lane&15, row m = v + 8*(lane>=16)).
    const float bias = b_rel[nBase + r];
#pragma unroll
    for (int v = 0; v < 8; ++v) c[v] += bias;

    // Fused mean-pool accumulation. batch is sorted, so a 16-node strip is
    // almost always one graph -> reduce rows in-register + one atomic/lane.
    const int g0  = batch[mBase];
    const int g15 = batch[mBase + 15];
    if (g0 == g15) {
        float s = 0.0f;
#pragma unroll
        for (int v = 0; v < 8; ++v) s += c[v];      // rows of this half-wave
        s += __shfl_xor(s, 16, 32);                 // combine the two halves
        if (lane < 16)
            atomicAdd(&summed[g0 * HIDDEN + nBase + lane], s);
    } else {
        const int mOff = (lane >> 4) * 8;
#pragma unroll
        for (int v = 0; v < 8; ++v) {
            const int m = mBase + v + mOff;
            atomicAdd(&summed[batch[m] * HIDDEN + nBase + r], c[v]);
        }
    }
}

// ---------------------------------------------------------------------------
// Final classifier: out[g][c] = (summed[g]/max(count,1)) . w_lin[c] + b_lin[c]
// 640 outputs -> single block.
// ---------------------------------------------------------------------------
__global__ void classify_kernel(const float* __restrict__ summed,
                                const float* __restrict__ counts,
                                const float* __restrict__ w_lin,
                                const float* __restrict__ b_lin,
                                float* __restrict__ out) {
    for (int idx = threadIdx.x; idx < N_GRAPHS * N_CLASSES; idx += blockDim.x) {
        const int g = idx / N_CLASSES;
        const int cc = idx % N_CLASSES;
        const float inv = 1.0f / fmaxf(counts[g], 1.0f);
        float acc = b_lin[cc];
        for (int k = 0; k < HIDDEN; ++k)
            acc += summed[g * HIDDEN + k] * inv * w_lin[cc * HIDDEN + k];
        out[g * N_CLASSES + cc] = acc;
    }
}

// ---------------------------------------------------------------------------
extern "C" void kernel_launch(void* const* d_in, const int* in_sizes, int n_in,
                              void* d_out, int out_size, void* d_ws, size_t ws_size,
                              hipStream_t stream) {
    const float* x      = (const float*)d_in[0];
    const int*   ei     = (const int*)d_in[1];
    const int*   batch  = (const int*)d_in[2];
    const float* w_rel  = (const float*)d_in[3];
    const float* b_rel  = (const float*)d_in[4];
    const float* w_root = (const float*)d_in[5];
    const float* w_lin  = (const float*)d_in[6];
    const float* b_lin  = (const float*)d_in[7];
    float*       out    = (float*)d_out;

    // Workspace layout: [agg: N*F floats][summed: G*H floats][counts: G floats]
    float* agg    = (float*)d_ws;
    float* summed = agg + (size_t)N_NODES * D_FEAT;
    float* counts = summed + (size_t)N_GRAPHS * HIDDEN;
    const long zeroElems = (long)N_NODES * D_FEAT + N_GRAPHS * HIDDEN + N_GRAPHS;

    zero_ws_kernel<<<4096, 256, 0, stream>>>(agg, zeroElems);

    const int n_edges = in_sizes[1] / 2;
    const long sthreads = (long)n_edges * 32;
    scatter_add_kernel<<<(int)((sthreads + 255) / 256), 256, 0, stream>>>(
        x, ei, agg, n_edges);

    count_kernel<<<(N_NODES + 255) / 256, 256, 0, stream>>>(batch, counts, N_NODES);

    gemm_pool_kernel<<<N_NODES / 16, 256, 0, stream>>>(
        x, agg, batch, w_rel, b_rel, w_root, summed);

    classify_kernel<<<1, 256, 0, stream>>>(summed, counts, w_lin, b_lin, out);
}